// PhysicalCircleLayer_67138928771493
// MI455X (gfx1250) — compile-verified
//
#include <hip/hip_runtime.h>
#include <math.h>
#include <stdint.h>

#define NS_DIM   100
#define NPIX     (NS_DIM * NS_DIM)   // 10000
#define NF4      (NPIX / 4)          // 2500 float4 per batch row
#define F4_ROW   (NS_DIM / 4)        // 25 float4 per map row (exact)
#define PART     8
#define MAXP     16
#define BLK      256
#define NSTAGE   ((NF4 + BLK - 1) / BLK)  // 10
#define DEPTH    4                        // async pipeline depth (buffers)
#define INF_F    1000000000.0f
#define MU_F     1e-8f
#define TWO_PI_F 6.28318530717958647692f

#define WAIT_ASYNC(n) asm volatile("s_wait_asynccnt " #n ::: "memory")

__global__ __launch_bounds__(BLK) void PhysicalCircleLayer_kernel(
    const float* __restrict__ seg_maps,     // (B, 100, 100)
    const float* __restrict__ seg_map_paras,// (B, 6): W0 W1 b0 b1 order0 0
    const float* __restrict__ trajectories, // (B, 8, 2)
    const float* __restrict__ current_pos,  // (B, 1, 2)
    float* __restrict__ out)                // (B, 16, 3)
{
    const int b   = blockIdx.x;
    const int tid = threadIdx.x;

    // DEPTH-deep staging: each lane owns one 16B slot per buffer stage.
    __shared__ float4   buf[DEPTH][BLK];   // 16 KB
    __shared__ unsigned smin[PART];

    if (tid < PART) smin[tid] = __float_as_uint(INF_F);

    // ---- per-batch uniforms (block-uniform -> scalar loads/ALU) ----
    const float W0 = seg_map_paras[b * 6 + 0];
    const float W1 = seg_map_paras[b * 6 + 1];
    const float b0 = seg_map_paras[b * 6 + 2];
    const float b1 = seg_map_paras[b * 6 + 3];
    const bool  swp = (seg_map_paras[b * 6 + 4] == 1.0f);
    // current_pos cancels in moving_vector = _obs[-1] - _obs[0]
    const float mvx = trajectories[b * 16 + 14] - trajectories[b * 16 + 0];
    const float mvy = trajectories[b * 16 + 15] - trajectories[b * 16 + 1];
    const float ml  = __builtin_amdgcn_sqrtf(mvx * mvx + mvy * mvy);
    const float r2  = (2.0f * ml) * (2.0f * ml);   // (VISION_RADIUS * len)^2
    const float cx  = current_pos[b * 2 + 0];
    const float cy  = current_pos[b * 2 + 1];
    const float rW0 = 1.0f / W0;
    const float rW1 = 1.0f / W1;
    // component from mp0 is dy (no swap) or dx (swap); subtract matching center
    const float cFix = swp ? cy : cx;
    const float cVar = swp ? cx : cy;

    const unsigned long long gbase =
        (unsigned long long)(seg_maps + (size_t)b * NPIX);
    const unsigned ldsbase = (unsigned)(uintptr_t)(&buf[0][0]);

    // Async copy of one float4 per lane into this lane's private LDS slot.
    auto issue = [&](int s) {
        int g = s * BLK + tid;
        if (g > NF4 - 1) g = NF4 - 1;   // clamp tail: keeps per-wave ASYNCcnt exact
        const unsigned goff  = (unsigned)g * 16u;
        const unsigned laddr = ldsbase + (unsigned)((((s & (DEPTH-1)) * BLK) + tid) * 16);
        asm volatile("global_load_async_to_lds_b128 %0, %1, %2"
                     :: "v"(laddr), "v"(goff), "s"(gbase) : "memory");
    };

    float lmin[PART];
#pragma unroll
    for (int p = 0; p < PART; ++p) lmin[p] = INF_F;

    // Prologue: DEPTH-1 stages in flight before consuming.
    for (int s = 0; s < DEPTH - 1; ++s) issue(s);

    for (int s = 0; s < NSTAGE; ++s) {
        if (s + DEPTH - 1 < NSTAGE) issue(s + DEPTH - 1);
        // After optional issue, outstanding = min(NSTAGE, s+DEPTH) - s stages;
        // stage s is complete once ASYNCcnt <= outstanding-1.
        {
            const int w = (NSTAGE - 1 - s) < (DEPTH - 1) ? (NSTAGE - 1 - s)
                                                         : (DEPTH - 1);
            if      (w >= 3) WAIT_ASYNC(3);
            else if (w == 2) WAIT_ASYNC(2);
            else if (w == 1) WAIT_ASYNC(1);
            else             WAIT_ASYNC(0);
        }

        const int g = s * BLK + tid;
        if (g < NF4) {
            const float4 v = buf[s & (DEPTH-1)][tid];
            const float mv[4] = {v.x, v.y, v.z, v.w};

            // A float4 never crosses a map row (100/4 == 25): hoist row terms.
            const int   row  = g / F4_ROW;
            const int   colb = (g - row * F4_ROW) * 4;
            const float fix  = ((float)row  - b0) * rW0 - cFix; // fixed comp
            const float var0 = ((float)colb - b1) * rW1 - cVar; // + e*rW1
            const float f2   = fix * fix;

            float var[4], d2[4];
            bool  ok[4];
            var[0] = var0;
            var[1] = var0 + rW1;
            var[2] = var[1] + rW1;
            var[3] = var[2] + rW1;
#pragma unroll
            for (int e = 0; e < 4; ++e) {
                d2[e] = fmaf(var[e], var[e], f2);
                ok[e] = (mv[e] > 0.05f) && (d2[e] <= r2); // ~safe & in-radius
            }

            // ~2-6% of pixels lie in the vision disk: skip heavy math when the
            // whole wave's 128-pixel arc misses it (s_cbranch_execz).
            if (ok[0] || ok[1] || ok[2] || ok[3]) {
#pragma unroll
                for (int e = 0; e < 4; ++e) {
                    const float dist = __builtin_amdgcn_sqrtf(d2[e]);
                    const float den  = mv[e] + MU_F;
                    float       rc   = __builtin_amdgcn_rcpf(den);
                    rc = rc * (2.0f - den * rc);            // 1 Newton step
                    const float eq   = (dist + MU_F) * rc;
                    const float dval = ok[e] ? eq : INF_F;

                    const float dy = swp ? var[e] : fix;  // dir[0]
                    const float dx = swp ? fix : var[e];  // dir[1]
                    // Octant == floor(mod(atan2(dy,dx),2pi) / (pi/4))
                    const float ax = fabsf(dx), ay = fabsf(dy);
                    int bin;
                    if (dx >= 0.0f) {
                        if (dy >= 0.0f) bin = (ay <  ax) ? 0 : 1;  // [0,90)
                        else            bin = (ay <= ax) ? 7 : 6;  // [270,360)
                    } else {
                        if (dy >= 0.0f) bin = (ay <= ax) ? 3 : 2;  // [90,180)
                        else            bin = (ay <  ax) ? 4 : 5;  // [180,270)
                    }
#pragma unroll
                    for (int p = 0; p < PART; ++p)
                        lmin[p] = (bin == p) ? fminf(lmin[p], dval) : lmin[p];
                }
            }
        }
    }

    __syncthreads();  // smin init visible; all lanes done computing
#pragma unroll
    for (int p = 0; p < PART; ++p)
        atomicMin(&smin[p], __float_as_uint(lmin[p]));  // ds_min_u32; f32>=0 order-safe
    __syncthreads();

    if (tid < MAXP) {
        float o0 = 0.0f, o1 = 0.0f, o2 = 0.0f;
        if (tid < PART) {
            const float md = __uint_as_float(smin[tid]);
            if (md < INF_F) {
                o0 = ml;
                o1 = md;
                o2 = TWO_PI_F * ((float)tid + 0.5f) / (float)PART;
            }
        }
        float* op = out + (size_t)b * (MAXP * 3) + tid * 3;
        op[0] = o0; op[1] = o1; op[2] = o2;
    }
}

extern "C" void kernel_launch(void* const* d_in, const int* in_sizes, int n_in,
                              void* d_out, int out_size, void* d_ws, size_t ws_size,
                              hipStream_t stream) {
    const float* seg_maps      = (const float*)d_in[0];
    const float* seg_map_paras = (const float*)d_in[1];
    const float* trajectories  = (const float*)d_in[2];
    const float* current_pos   = (const float*)d_in[3];
    float*       out           = (float*)d_out;

    const int Bn = in_sizes[1] / 6;   // seg_map_paras is (B, 6)

    PhysicalCircleLayer_kernel<<<Bn, BLK, 0, stream>>>(
        seg_maps, seg_map_paras, trajectories, current_pos, out);
}